// BayesianSparseLinear_41197326303442
// MI455X (gfx1250) — compile-verified
//
#include <hip/hip_runtime.h>

#define G1 256
#define G2 256
#define A1 32
#define A2 32
#define E_EDGES 4096
#define BATCH 1024
#define BN 128          // batch-tile width per workgroup
#define WSTRIDE 36      // padded row stride (floats) for W^T in LDS
#define XSTRIDE 144     // padded row stride (floats) for x tile in LDS

typedef float v2f __attribute__((ext_vector_type(2)));
typedef float v4f __attribute__((ext_vector_type(4)));
typedef float v8f __attribute__((ext_vector_type(8)));
typedef int   v4i __attribute__((ext_vector_type(4)));

#if __has_builtin(__builtin_amdgcn_global_load_async_to_lds_b128)
#define HAVE_ASYNC_LDS 1
typedef __attribute__((address_space(1))) v4i as1_v4i;   // global int4
typedef __attribute__((address_space(3))) v4i as3_v4i;   // LDS int4
#else
#define HAVE_ASYNC_LDS 0
#endif

static __device__ __forceinline__ void wait_asynccnt0() {
#if __has_builtin(__builtin_amdgcn_s_wait_asynccnt)
    __builtin_amdgcn_s_wait_asynccnt(0);
#else
    asm volatile("s_wait_asynccnt 0x0" ::: "memory");
#endif
}

// One workgroup per (output graph node g, 128-wide batch tile). 256 threads =
// 8 waves; wave w owns columns [w*16, w*16+16) and all 32 output rows (two
// 16x16 f32 WMMA accumulators). Edges incident to g are processed in
// ascending index order (deterministic), accumulating in registers (no
// atomics). Per-edge staging is double-buffered: edge t+1's global traffic
// (async DMA to LDS for x, register fetch for weights) is issued before the
// WMMA burst for edge t, hiding global latency under matrix work.
__global__ __launch_bounds__(256) void bayes_block_spmm_kernel(
    const float* __restrict__ x,          // [G1*A1, B]
    const int*   __restrict__ row_g,      // [E]
    const int*   __restrict__ col_g,      // [E]
    const float* __restrict__ w_mean,     // [E, A1, A2]
    const float* __restrict__ w_lv,       // [E, A1, A2]
    const float* __restrict__ w_eps,      // [E, A1, A2]
    const float* __restrict__ b_mean,     // [G2*A2]
    const float* __restrict__ b_lv,       // [G2*A2]
    const float* __restrict__ b_eps,      // [G2*A2]
    float*       __restrict__ out)        // [G2*A2, B] + 1 (kl)
{
    __shared__ float wlds[2][A2 * WSTRIDE];   // W[j][i] = v[i][j], padded
    __shared__ float xlds[2][A1 * XSTRIDE];   // x block tile [i][b], padded
    __shared__ int   s_cnt[256];
    __shared__ int   s_list[E_EDGES];

    const int g   = blockIdx.x;            // output graph node
    const int b0  = blockIdx.y * BN;       // batch tile start
    const int tid = threadIdx.x;

    if (blockIdx.x == 0 && blockIdx.y == 0 && tid == 0) {
        out[(size_t)G2 * A2 * BATCH] = 0.0f;   // kl scalar
    }

    // ---- Phase 0: deterministic, index-sorted list of edges with row_g==g --
    const int ebase = tid * (E_EDGES / 256);
    int cnt = 0;
    #pragma unroll
    for (int q = 0; q < E_EDGES / 256; ++q)
        cnt += (row_g[ebase + q] == g) ? 1 : 0;
    s_cnt[tid] = cnt;
    __syncthreads();
    for (int off = 1; off < 256; off <<= 1) {     // Hillis-Steele inclusive scan
        int add = (tid >= off) ? s_cnt[tid - off] : 0;
        __syncthreads();
        s_cnt[tid] += add;
        __syncthreads();
    }
    int pos = s_cnt[tid] - cnt;                   // exclusive prefix
    const int total = s_cnt[255];
    for (int q = 0; q < E_EDGES / 256; ++q) {
        int e = ebase + q;
        if (row_g[e] == g) s_list[pos++] = e;
    }
    __syncthreads();

    // ---- WMMA lane decomposition ------------------------------------------
    const int wave = tid >> 5;
    const int lane = tid & 31;
    const int hh   = lane >> 4;      // half-wave: selects K pair of A/B frags
    const int lm   = lane & 15;      // M index (A) / N index (B)
    const int n0   = wave * 16;      // this wave's column offset in the tile

    v8f c0 = {};                     // output rows 0..15 of the block
    v8f c1 = {};                     // output rows 16..31 of the block

    // ---- Staging registers for software pipeline --------------------------
    float wm[4], wl[4], we[4];
#if !HAVE_ASYNC_LDS
    v4f xreg[4];
#endif

    // Issue all global traffic for edge e (weights -> regs, x -> async DMA
    // straight into LDS buffer `buf`, tracked by ASYNCcnt).
    auto fetch_edge = [&](int e, int buf) {
        const size_t wbase = (size_t)e * (A1 * A2);
        #pragma unroll
        for (int p = 0; p < 4; ++p) {
            int idx = p * 256 + tid;
            wm[p] = w_mean[wbase + idx];
            wl[p] = w_lv[wbase + idx];
            we[p] = w_eps[wbase + idx];
        }
        const int cg = col_g[e];
        const float* xsrc = x + (size_t)cg * (A1 * BATCH) + b0;
        #pragma unroll
        for (int p = 0; p < 4; ++p) {
            int idx = p * 256 + tid;              // float4 index: 32 rows x 32
            int i = idx >> 5, c4 = idx & 31;
#if HAVE_ASYNC_LDS
            __builtin_amdgcn_global_load_async_to_lds_b128(
                (as1_v4i*)(xsrc + (size_t)i * BATCH + c4 * 4),
                (as3_v4i*)&xlds[buf][i * XSTRIDE + c4 * 4],
                0, 0);
#else
            xreg[p] = *(const v4f*)(xsrc + (size_t)i * BATCH + c4 * 4);
#endif
        }
    };

    // Finish staging into buffer `buf`: sample weights (transposed) into LDS,
    // land the x tile, and drain ASYNCcnt for this wave's DMA.
    auto commit_edge = [&](int buf) {
        #pragma unroll
        for (int p = 0; p < 4; ++p) {
            int idx = p * 256 + tid;              // idx = i*32 + j
            int i = idx >> 5, j = idx & 31;
            wlds[buf][j * WSTRIDE + i] = fmaf(we[p], expf(wl[p]), wm[p]);
        }
#if HAVE_ASYNC_LDS
        wait_asynccnt0();
#else
        #pragma unroll
        for (int p = 0; p < 4; ++p) {
            int idx = p * 256 + tid;
            int i = idx >> 5, c4 = idx & 31;
            *(v4f*)&xlds[buf][i * XSTRIDE + c4 * 4] = xreg[p];
        }
#endif
    };

    // 32x128 tile matmul from LDS buffer `buf` via f32 WMMA, K steps of 4.
    // A 16x4: lanes 0-15 -> M=lm, K={k0,k0+1}; lanes 16-31 -> K={k0+2,k0+3}.
    // B 4x16 mirrors it (N=lm). C/D: VGPR r -> M=r / r+8; N = lane&15.
    auto compute_tile = [&](int buf) {
        #pragma unroll
        for (int k0 = 0; k0 < A1; k0 += 4) {
            int ka = k0 + hh * 2;
            v2f a0 = *(const v2f*)&wlds[buf][lm * WSTRIDE + ka];
            v2f a1 = *(const v2f*)&wlds[buf][(lm + 16) * WSTRIDE + ka];
            v2f bf;
            bf.x = xlds[buf][ka * XSTRIDE + n0 + lm];
            bf.y = xlds[buf][(ka + 1) * XSTRIDE + n0 + lm];
            c0 = __builtin_amdgcn_wmma_f32_16x16x4_f32(
                     false, a0, false, bf, (short)0, c0, false, false);
            c1 = __builtin_amdgcn_wmma_f32_16x16x4_f32(
                     false, a1, false, bf, (short)0, c1, false, false);
        }
    };

    // ---- Pipelined main loop over incident edges --------------------------
    if (total > 0) {
        fetch_edge(s_list[0], 0);
        commit_edge(0);
    }
    __syncthreads();
    for (int t = 0; t < total; ++t) {
        const int cur = t & 1;
        if (t + 1 < total) fetch_edge(s_list[t + 1], cur ^ 1); // overlap t+1
        compute_tile(cur);
        if (t + 1 < total) commit_edge(cur ^ 1);
        __syncthreads();   // separates buffer roles between iterations
    }

    // ---- Epilogue: add sampled bias, store once (no atomics) --------------
    const int n = b0 + n0 + lm;
    #pragma unroll
    for (int r = 0; r < 8; ++r) {
        int j0   = hh * 8 + r;
        int row0 = g * A2 + j0;
        int row1 = row0 + 16;
        float bias0 = fmaf(b_eps[row0], expf(b_lv[row0]), b_mean[row0]);
        float bias1 = fmaf(b_eps[row1], expf(b_lv[row1]), b_mean[row1]);
        out[(size_t)row0 * BATCH + n] = c0[r] + bias0;
        out[(size_t)row1 * BATCH + n] = c1[r] + bias1;
    }
}

extern "C" void kernel_launch(void* const* d_in, const int* in_sizes, int n_in,
                              void* d_out, int out_size, void* d_ws, size_t ws_size,
                              hipStream_t stream) {
    (void)in_sizes; (void)n_in; (void)out_size; (void)d_ws; (void)ws_size;
    const float* x      = (const float*)d_in[0];
    const int*   row_g  = (const int*)  d_in[1];
    const int*   col_g  = (const int*)  d_in[2];
    const float* w_mean = (const float*)d_in[3];
    const float* w_lv   = (const float*)d_in[4];
    const float* w_eps  = (const float*)d_in[5];
    const float* b_mean = (const float*)d_in[6];
    const float* b_lv   = (const float*)d_in[7];
    const float* b_eps  = (const float*)d_in[8];
    float*       out    = (float*)d_out;

    dim3 grid(G2, BATCH / BN);   // 256 x 8 workgroups
    bayes_block_spmm_kernel<<<grid, 256, 0, stream>>>(
        x, row_g, col_g, w_mean, w_lv, w_eps, b_mean, b_lv, b_eps, out);
}